// Conv2DUF_32847909879889
// MI455X (gfx1250) — compile-verified
//
#include <hip/hip_runtime.h>
#include <stdint.h>

// ---------------------------------------------------------------------------
// Conv 3x3 (N=8, Cin=16, Cout=16, 512x512, stride 1, pad 1) as im2col GEMM
// on CDNA5 fp32 WMMA.
//  - 4 output rows x 128 pixels per block; 6 input rows staged in LDS via
//    global_load_async_to_lds_b128 (uniform per-chunk validity, zero-fill pads)
//  - inner loop per K-group: 2x ds_load_b32 (imm offset) + v_wmma_f32_16x16x4_f32
//  - K-permutation: lane-half 0 -> k 0..71 (ci 0..7), lane-half 1 -> k 72..143
// ---------------------------------------------------------------------------

typedef float v2f __attribute__((ext_vector_type(2)));
typedef float v4f __attribute__((ext_vector_type(4)));
typedef float v8f __attribute__((ext_vector_type(8)));

#define CO 16
#define CI 16
#define HH 512
#define WW 512
#define KDIM 144
#define NG 36
#define BW   128                    // output pixels per block
#define ROWS 4                      // output rows per block
#define SROWS 6                     // staged input rows (ROWS + 2)
#define LSTRIDE 140                 // words per staged row (136 used, 4-aligned)
#define LCH (SROWS * LSTRIDE + 4)   // 844 words; 8*LCH % 64 == 32 (bank split)
#define LDS_WORDS (CI * LCH)        // 13504 words = 52.75 KB
#define CHUNKS_ROW (LSTRIDE / 4)    // 35 x b128 chunks per row
#define CHUNKS (CI * SROWS * CHUNKS_ROW)  // 3360

// k in 0..71: LDS word offset (ci = k/9, kh = (k%9)/3, kw = k%3)
__device__ constexpr int loff(int k) {
    return (k / 9) * LCH + ((k % 9) / 3) * LSTRIDE + (k % 3);
}

__global__ __launch_bounds__(256)
void conv3x3_wmma_f32(const float* __restrict__ x,
                      const float* __restrict__ wgt,
                      float* __restrict__ out)
{
    __shared__ float smem[LDS_WORDS];

    const int tid  = threadIdx.x;
    const int lane = tid & 31;
    const int wave = tid >> 5;           // 0..7
    const int wblk = blockIdx.x * BW;
    const int h0   = blockIdx.y * ROWS;
    const int n    = blockIdx.z;
    const int half = lane >> 4;          // 0: k 0..71, 1: k 72..143
    const int mrow = lane & 15;          // A row (co) / D column (px)

    // ---- stage input rows h0-1 .. h0+4, cols wblk-4 .. wblk+135 ------------
    // LDS col c <-> global col wblk-4+c ; every b128 chunk is uniformly
    // valid or invalid (boundaries are multiples of 4).
    {
        const float* xb = x + (size_t)n * CI * HH * WW;
        const uint32_t lds_base = (uint32_t)(uintptr_t)&smem[0];
        for (int i4 = tid; i4 < CHUNKS; i4 += 256) {
            int ci  = i4 / (SROWS * CHUNKS_ROW);
            int rem = i4 - ci * (SROWS * CHUNKS_ROW);
            int r   = rem / CHUNKS_ROW;
            int c4  = rem - r * CHUNKS_ROW;
            int gw0 = wblk - 4 + 4 * c4;         // first global col of chunk
            int hr  = h0 - 1 + r;                // global input row
            int widx = ci * LCH + r * LSTRIDE + 4 * c4;
            bool ok = (gw0 >= 0) && (gw0 < WW) && (hr >= 0) && (hr < HH);
            if (ok) {
                const float* gp = xb + ((size_t)ci * HH + hr) * WW + gw0;
                uint32_t la = lds_base + widx * 4;
                asm volatile("global_load_async_to_lds_b128 %0, %1, off"
                             :: "v"(la), "v"(gp) : "memory");
            } else {
                v4f z = {0.0f, 0.0f, 0.0f, 0.0f};
                *(v4f*)&smem[widx] = z;          // zero padding chunk
            }
        }
    }

    // ---- A fragments: contiguous 72-float slab per lane (overlaps async) ---
    v2f a[NG];
    {
        const float* wrow = wgt + mrow * KDIM + half * 72;
#pragma unroll
        for (int g = 0; g < NG; ++g)
            a[g] = *(const v2f*)(wrow + 2 * g);
    }

    asm volatile("s_wait_asynccnt 0x0" ::: "memory");
    __syncthreads();

    // ---- compute: each wave owns a 16-px column strip across 4 rows --------
    const int px = wave * 16 + mrow;     // block-local pixel column
#pragma unroll
    for (int r = 0; r < ROWS; ++r) {
        // base LDS col for this lane: c = px + 4 + (kw-1) = px + 3 + kw
        const float* sb = smem + half * (8 * LCH) + r * LSTRIDE + px + 3;

        v8f acc = {};
#pragma unroll
        for (int g = 0; g < NG; ++g) {
            v2f b;
            b.x = sb[loff(2 * g)];        // ds_load_b32, immediate offset
            b.y = sb[loff(2 * g + 1)];    // ds_load_b32, immediate offset
            acc = __builtin_amdgcn_wmma_f32_16x16x4_f32(
                /*neg_a=*/false, a[g],
                /*neg_b=*/false, b,
                /*c_mod=*/(short)0, acc,
                /*reuse_a=*/false, /*reuse_b=*/false);
        }

        // D: VGPR v holds co = v + 8*half, column = mrow
        float* ob = out + (((size_t)n * CO + half * 8) * HH + (h0 + r)) * WW
                  + wblk + px;
#pragma unroll
        for (int v = 0; v < 8; ++v)
            ob[(size_t)v * HH * WW] = acc[v];
    }
}

extern "C" void kernel_launch(void* const* d_in, const int* in_sizes, int n_in,
                              void* d_out, int out_size, void* d_ws, size_t ws_size,
                              hipStream_t stream) {
    const float* x   = (const float*)d_in[0];   // [8,16,512,512] fp32
    const float* wgt = (const float*)d_in[1];   // [16,144] fp32
    float*       out = (float*)d_out;           // [8,16,512,512] fp32

    dim3 grid(WW / BW, HH / ROWS, 8);   // (w-chunk, h-chunk, n) = (4,128,8)
    dim3 block(256, 1, 1);              // 8 wave32 waves
    conv3x3_wmma_f32<<<grid, block, 0, stream>>>(x, wgt, out);
}